// GraphConv_59828894433559
// MI455X (gfx1250) — compile-verified
//
#include <hip/hip_runtime.h>

// GraphConv: out = segment_sum(edge_val * (x@W)[edge_src] -> edge_dst) + b
//
// Phase 1: xw = x @ W   (WMMA f32 16x16x4; 80-row blocks for W reuse;
//                        async global->LDS staging of the x tile)     -> d_ws
// Phase 2: out[n][:] = b[:]                                           (float4 fill)
// Phase 3: out[dst][:] += val * xw[src][:]  (wave-per-edge, f32 atomics, L2-resident)

typedef float v2f __attribute__((ext_vector_type(2)));
typedef float v8f __attribute__((ext_vector_type(8)));
typedef int   i32x4 __attribute__((vector_size(16)));   // matches builtin param type

#define IN_CH 256
#define OUT_CH 256
#define LDS_STRIDE 260        // pad rows to 260 floats -> conflict-free ds_load_b64
#define ROW_TILES 5           // 5 x 16 = 80 rows per block; 50000 = 625 * 80 exact
#define ROWS_PER_BLOCK (ROW_TILES * 16)

#if defined(__has_builtin)
#if __has_builtin(__builtin_amdgcn_global_load_async_to_lds_b128) && \
    __has_builtin(__builtin_amdgcn_s_wait_asynccnt)
#define USE_ASYNC_LDS 1
#endif
#endif

// ---------------------------------------------------------------------------
// Phase 1: xw[r][c] = sum_k x[r][k] * W[k][c]
// Block = 256 threads = 8 waves; 80-row stripe per block.
// Wave w computes col tiles c0 = 16*w, c1 = c0+128 for all 5 row tiles:
// B fragments reused 5x, A fragments reused 2x -> W L2 traffic 160 MB total.
// ---------------------------------------------------------------------------
__global__ void __launch_bounds__(256)
gemm_xw_kernel(const float* __restrict__ x, const float* __restrict__ W,
               float* __restrict__ xw) {
    __shared__ float xs[ROWS_PER_BLOCK * LDS_STRIDE];   // 83,200 B

    const int r0  = blockIdx.x * ROWS_PER_BLOCK;
    const int tid = threadIdx.x;

    // Stage 80x256 f32 (5120 float4) with 256 threads -> 20 float4 each.
#pragma unroll
    for (int it = 0; it < 20; ++it) {
        const int idx  = it * 256 + tid;   // float4 index in tile
        const int row  = idx >> 6;         // 64 float4 per row
        const int col4 = idx & 63;
        const float* gp = x + (size_t)(r0 + row) * IN_CH + col4 * 4;
        float* lp = &xs[row * LDS_STRIDE + col4 * 4];
#if USE_ASYNC_LDS
        __builtin_amdgcn_global_load_async_to_lds_b128(
            (__attribute__((address_space(1))) i32x4*)gp,
            (__attribute__((address_space(3))) i32x4*)lp, 0, 0);
#else
        const float4 v = *(const float4*)gp;
        lp[0] = v.x; lp[1] = v.y; lp[2] = v.z; lp[3] = v.w;
#endif
    }
#if USE_ASYNC_LDS
    __builtin_amdgcn_s_wait_asynccnt(0);
#endif
    __syncthreads();

    const int wave  = tid >> 5;
    const int lane  = tid & 31;
    const int m     = lane & 15;   // A row / B column / D column within tile
    const int khalf = lane >> 4;   // 0 -> K {0,1}, 1 -> K {2,3}
    const int c0    = wave * 16;
    const int c1    = c0 + 128;

    v8f acc[ROW_TILES][2];
#pragma unroll
    for (int t = 0; t < ROW_TILES; ++t) { acc[t][0] = (v8f){}; acc[t][1] = (v8f){}; }

#pragma unroll 2
    for (int k0 = 0; k0 < IN_CH; k0 += 4) {
        const int kA = k0 + 2 * khalf;
        const float* wrow = W + (size_t)kA * OUT_CH + m;
        v2f b0, b1;
        b0.x = wrow[c0];
        b0.y = wrow[OUT_CH + c0];
        b1.x = wrow[c1];
        b1.y = wrow[OUT_CH + c1];

#pragma unroll
        for (int t = 0; t < ROW_TILES; ++t) {
            const v2f a = *(const v2f*)&xs[(16 * t + m) * LDS_STRIDE + 2 * khalf + k0];
            acc[t][0] = __builtin_amdgcn_wmma_f32_16x16x4_f32(false, a, false, b0,
                                                              (short)0, acc[t][0], false, false);
            acc[t][1] = __builtin_amdgcn_wmma_f32_16x16x4_f32(false, a, false, b1,
                                                              (short)0, acc[t][1], false, false);
        }
    }

    // D layout: VGPR i, lanes 0-15 -> row i, lanes 16-31 -> row i+8; col = lane&15
#pragma unroll
    for (int t = 0; t < ROW_TILES; ++t) {
        const int outRow = r0 + 16 * t + khalf * 8;
#pragma unroll
        for (int i = 0; i < 8; ++i) {
            xw[(size_t)(outRow + i) * OUT_CH + c0 + m] = acc[t][0][i];
            xw[(size_t)(outRow + i) * OUT_CH + c1 + m] = acc[t][1][i];
        }
    }
}

// ---------------------------------------------------------------------------
// Phase 2: out[n][c] = b[c]  (broadcast rows; atomics accumulate on top)
// ---------------------------------------------------------------------------
__global__ void __launch_bounds__(256)
bias_init_kernel(const float* __restrict__ b, float* __restrict__ out) {
    const int i = blockIdx.x * 256 + threadIdx.x;       // float4 index
    const float4 bb = ((const float4*)b)[i & (OUT_CH / 4 - 1)];
    ((float4*)out)[i] = bb;
}

// ---------------------------------------------------------------------------
// Phase 3: one wave per edge; lane owns 8 contiguous channels.
// Gathers: 2x b128 per lane; scatter: 8 global_atomic_add_f32 per lane.
// xw and out are both L2-resident (51.2 MB each, 192 MB L2).
// ---------------------------------------------------------------------------
__global__ void __launch_bounds__(256)
spmm_scatter_kernel(const float* __restrict__ xw,
                    const int* __restrict__ esrc,
                    const int* __restrict__ edst,
                    const float* __restrict__ eval,
                    float* __restrict__ out, int n_edges) {
    const int wave = (blockIdx.x * 256 + threadIdx.x) >> 5;
    const int lane = threadIdx.x & 31;
    if (wave >= n_edges) return;

    const int   s = __builtin_amdgcn_readfirstlane(esrc[wave]);
    const int   d = __builtin_amdgcn_readfirstlane(edst[wave]);
    const float v = eval[wave];

    const float4* xp = (const float4*)(xw + (size_t)s * OUT_CH) + lane * 2;
    const float4 m0 = xp[0];
    const float4 m1 = xp[1];

    float* op = out + (size_t)d * OUT_CH + lane * 8;
    atomicAdd(op + 0, v * m0.x);
    atomicAdd(op + 1, v * m0.y);
    atomicAdd(op + 2, v * m0.z);
    atomicAdd(op + 3, v * m0.w);
    atomicAdd(op + 4, v * m1.x);
    atomicAdd(op + 5, v * m1.y);
    atomicAdd(op + 6, v * m1.z);
    atomicAdd(op + 7, v * m1.w);
}

// ---------------------------------------------------------------------------
extern "C" void kernel_launch(void* const* d_in, const int* in_sizes, int n_in,
                              void* d_out, int out_size, void* d_ws, size_t ws_size,
                              hipStream_t stream) {
    const float* x    = (const float*)d_in[0];
    const float* W    = (const float*)d_in[1];
    const float* b    = (const float*)d_in[2];
    const int*   esrc = (const int*)d_in[3];
    const int*   edst = (const int*)d_in[4];
    const float* eval = (const float*)d_in[5];
    float*       out  = (float*)d_out;
    float*       xw   = (float*)d_ws;        // needs N_NODES*OUT_CH*4 = 51.2 MB

    const int n_nodes = in_sizes[0] / IN_CH; // 50000
    const int n_edges = in_sizes[3];         // 800000

    // Phase 1: dense transform (50000 = 625 * 80, exact)
    gemm_xw_kernel<<<n_nodes / ROWS_PER_BLOCK, 256, 0, stream>>>(x, W, xw);

    // Phase 2: bias broadcast init of out
    const int n_f4 = n_nodes * (OUT_CH / 4);
    bias_init_kernel<<<n_f4 / 256, 256, 0, stream>>>(b, out);

    // Phase 3: edge scatter (8 waves per 256-thread block)
    const int blocks = (n_edges + 7) / 8;
    spmm_scatter_kernel<<<blocks, 256, 0, stream>>>(xw, esrc, edst, eval, out, n_edges);
}